// GCN_79568564125697
// MI455X (gfx1250) — compile-verified
//
#include <hip/hip_runtime.h>

typedef __attribute__((ext_vector_type(2))) float v2f;
typedef __attribute__((ext_vector_type(8))) float v8f;

#define NNODES 50000
#define NFEAT  256
#define NHID   128
#define NCLASS 32

// ---------------------------------------------------------------------------
// Register-blocked dense GEMM via V_WMMA_F32_16X16X4_F32:
//   out[M,N] = A[M,K] @ W[K,N] + bias
// One wave computes a 16 x (NT*16) strip: the A fragment (float2/lane) is
// loaded once per k-step and reused across NT independent WMMA accumulators,
// cutting A-side L2 traffic by NT and giving NT-way WMMA ILP per wave.
//
// Fragment layouts (ISA 7.12.2, 32-bit operands):
//   A 16x4 : lanes 0-15 -> K=k,k+1 of row M=lane; lanes 16-31 -> K=k+2,k+3
//            of row M=lane-16 (one contiguous float2 per lane).
//   B 4x16 : lanes 0-15 hold rows K=k,k+1 (cols n0+lane) across the 2 VGPRs;
//            lanes 16-31 hold rows K=k+2,k+3 (coalesced 16-wide loads).
//   C/D    : VGPR i -> row m0 + i + 8*(lane>=16), col n0 + (lane&15).
// ---------------------------------------------------------------------------
template <int KDIM, int NDIM, int NT, bool RELU>
__global__ void __launch_bounds__(32)
gcn_gemm_wmma(const float* __restrict__ A,     // [NNODES, KDIM] row-major
              const float* __restrict__ W,     // [KDIM, NDIM]   row-major
              const float* __restrict__ bias,  // [NDIM]
              float* __restrict__ out)         // [NNODES, NDIM]
{
    const int m0   = blockIdx.x << 4;
    const int n0   = blockIdx.y * (NT * 16);
    const int lane = threadIdx.x;
    const int half = lane >> 4;   // 0 = lanes 0-15, 1 = lanes 16-31
    const int l    = lane & 15;

    const float* arow  = A + (size_t)(m0 + l) * KDIM + (half << 1);
    const float* wbase = W + (size_t)(half << 1) * NDIM + (n0 + l);

    v8f acc[NT];
#pragma unroll
    for (int t = 0; t < NT; ++t) acc[t] = (v8f){};

#pragma unroll 2
    for (int k = 0; k < KDIM; k += 4) {
        float a0 = arow[k];
        float a1 = arow[k + 1];
        if (RELU) { a0 = fmaxf(a0, 0.0f); a1 = fmaxf(a1, 0.0f); }
        v2f a; a.x = a0; a.y = a1;
#pragma unroll
        for (int t = 0; t < NT; ++t) {
            v2f b;
            b.x = wbase[(size_t)k * NDIM + t * 16];
            b.y = wbase[(size_t)(k + 1) * NDIM + t * 16];
            // 8 args: (neg_a, A, neg_b, B, c_mod, C, reuse_a, reuse_b)
            acc[t] = __builtin_amdgcn_wmma_f32_16x16x4_f32(
                false, a, false, b, (short)0, acc[t], false, false);
        }
    }

#pragma unroll
    for (int t = 0; t < NT; ++t) {
        const float bv = bias[n0 + t * 16 + l];
        float* orow = out + (size_t)(m0 + (half << 3)) * NDIM + (n0 + t * 16 + l);
#pragma unroll
        for (int i = 0; i < 8; ++i)
            orow[(size_t)i * NDIM] = acc[t][i] + bv;
    }
}

// ---------------------------------------------------------------------------
// Weighted scatter-add SpMM: out[dst] += w * h[src].
// Consecutive threads cover consecutive float4 chunks of one edge's feature
// row: coalesced gather from h[src] and coalesced global_atomic_add_f32
// bursts into out[dst] (all L2-resident: 25 MB features vs 192 MB L2).
// ---------------------------------------------------------------------------
template <int F>
__global__ void __launch_bounds__(256)
gcn_spmm_atomic(const int* __restrict__ src, const int* __restrict__ dst,
                const float* __restrict__ w, const float* __restrict__ h,
                float* __restrict__ out, int nEdges)
{
    constexpr int CH = F / 4;  // float4 chunks per edge
    long long tid = (long long)blockIdx.x * blockDim.x + threadIdx.x;
    if (tid >= (long long)nEdges * CH) return;
    const int e = (int)(tid / CH);
    const int c = (int)(tid % CH);

    const float wt = w[e];
    const int   s  = src[e];
    const int   d  = dst[e];

    const float4 hv = *((const float4*)(h + (size_t)s * F) + c);
    float* o = out + (size_t)d * F + ((size_t)c << 2);
    atomicAdd(o + 0, wt * hv.x);
    atomicAdd(o + 1, wt * hv.y);
    atomicAdd(o + 2, wt * hv.z);
    atomicAdd(o + 3, wt * hv.w);
}

__global__ void __launch_bounds__(256)
zero_f32(float* __restrict__ p, long long n)
{
    long long i = (long long)blockIdx.x * blockDim.x + threadIdx.x;
    if (i < n) p[i] = 0.0f;
}

extern "C" void kernel_launch(void* const* d_in, const int* in_sizes, int n_in,
                              void* d_out, int out_size, void* d_ws, size_t ws_size,
                              hipStream_t stream)
{
    const float* x    = (const float*)d_in[0];
    const int*   esrc = (const int*)  d_in[1];
    const int*   edst = (const int*)  d_in[2];
    const float* ew   = (const float*)d_in[3];
    const float* W1   = (const float*)d_in[4];
    const float* b1   = (const float*)d_in[5];
    const float* W2   = (const float*)d_in[6];
    const float* b2   = (const float*)d_in[7];
    float*       outp = (float*)d_out;

    const int nE = in_sizes[1];

    // Workspace partition (floats): h1 | h1agg | h2
    float* h1    = (float*)d_ws;
    float* h1agg = h1    + (size_t)NNODES * NHID;
    float* h2    = h1agg + (size_t)NNODES * NHID;

    // 0) zero the SpMM accumulator
    {
        long long n = (long long)NNODES * NHID;
        zero_f32<<<(int)((n + 255) / 256), 256, 0, stream>>>(h1agg, n);
    }

    // 1) h1 = x @ W1 + b1   (50000 = 3125*16, no tail; NT=4 -> grid.y = 2)
    {
        dim3 grid(NNODES / 16, NHID / (16 * 4));
        gcn_gemm_wmma<NFEAT, NHID, 4, false><<<grid, 32, 0, stream>>>(x, W1, b1, h1);
    }

    // 2) h1agg[dst] += w * h1[src]
    {
        long long work = (long long)nE * (NHID / 4);
        gcn_spmm_atomic<NHID><<<(int)((work + 255) / 256), 256, 0, stream>>>(
            esrc, edst, ew, h1, h1agg, nE);
    }

    // 3) h2 = relu(h1agg) @ W2 + b2   (ReLU fused; NT=2 -> grid.y = 1)
    {
        dim3 grid(NNODES / 16, NCLASS / (16 * 2));
        gcn_gemm_wmma<NHID, NCLASS, 2, true><<<grid, 32, 0, stream>>>(h1agg, W2, b2, h2);
    }

    // 4) zero the output, then out[dst] += w * h2[src]
    {
        long long n = (long long)NNODES * NCLASS;
        zero_f32<<<(int)((n + 255) / 256), 256, 0, stream>>>(outp, n);
    }
    {
        long long work = (long long)nE * (NCLASS / 4);
        gcn_spmm_atomic<NCLASS><<<(int)((work + 255) / 256), 256, 0, stream>>>(
            esrc, edst, ew, h2, outp, nE);
    }
}